// QueryAndGroup_22505628631249
// MI455X (gfx1250) — compile-verified
//
#include <hip/hip_runtime.h>

typedef __attribute__((ext_vector_type(2))) float v2f;
typedef __attribute__((ext_vector_type(8))) float v8f;

constexpr int   B_   = 8;
constexpr int   N_   = 8192;
constexpr int   NP_  = 2048;
constexpr int   C_   = 64;
constexpr int   NS_  = 32;
constexpr float R2_  = 0.2f * 0.2f;

__device__ __forceinline__ float lane_bcast_f32(float v, int srcLane) {
    int r = __builtin_amdgcn_ds_bpermute(srcLane << 2, __builtin_bit_cast(int, v));
    return __builtin_bit_cast(float, r);
}

// ---------------------------------------------------------------------------
// Kernel A: pack each point as (-2x, -2y, -2z, |p|^2) so the WMMA B-operand is
// a single b64 load per lane in the hot loop.
// ---------------------------------------------------------------------------
__global__ __launch_bounds__(256) void pack_points_kernel(
    const float* __restrict__ xyz, float4* __restrict__ pp) {
    int i = blockIdx.x * blockDim.x + threadIdx.x;
    if (i >= B_ * N_) return;
    float x = xyz[3 * i + 0], y = xyz[3 * i + 1], z = xyz[3 * i + 2];
    pp[i] = make_float4(-2.f * x, -2.f * y, -2.f * z, x * x + y * y + z * z);
}

// ---------------------------------------------------------------------------
// Kernel B: ball query. One wave owns 16 query points (WMMA M dim) and sweeps
// the 8192 points in 16-column tiles. Each V_WMMA_F32_16X16X4_F32 yields a
// 16x16 tile of exact squared distances:
//   A row m = (qx, qy, qz, 1)            (lanes0-15: K0,K1; lanes16-31: K2,K3)
//   B col n = (-2px, -2py, -2pz, |p|^2)
//   C       = |q_m|^2 broadcast per row
//   D       = |q|^2 - 2 q.p + |p|^2 = |q - p|^2
// Double-buffered: tile k+1's b64 load is in flight while tile k runs through
// the WMMA + ballot + ordered selection, hiding VMEM latency.
// ---------------------------------------------------------------------------
__global__ __launch_bounds__(128) void ball_query_wmma_kernel(
    const float*  __restrict__ new_xyz,
    const float2* __restrict__ ppv,       // packed points viewed as float2 pairs
    int*          __restrict__ idxbuf) {
    const int lane = threadIdx.x & 31;
    const int wave = threadIdx.x >> 5;
    const int b    = blockIdx.y;
    const int q0   = blockIdx.x * 64 + wave * 16;
    const int half = (lane >= 16) ? 1 : 0;
    const int qIdx = q0 + (lane & 15);

    // ---- A operand + per-row |q|^2 accumulator ----
    const float* q = new_xyz + ((size_t)b * NP_ + qIdx) * 3;
    float qx = q[0], qy = q[1], qz = q[2];
    v2f a;
    a.x = half ? qz : qx;
    a.y = half ? 1.0f : qy;
    float qn2 = qx * qx + qy * qy + qz * qz;
    v8f cacc;
#pragma unroll
    for (int r = 0; r < 8; ++r)
        cacc[r] = lane_bcast_f32(qn2, r + (half ? 8 : 0));   // M = r (+8 hi half)

    // Per-lane element offset within a tile: lanes0-15 take float2 #0 of their
    // column (K0,K1 = -2px,-2py), lanes16-31 take float2 #1 (K2,K3 = -2pz,|p|^2).
    const float2* pl = ppv + (size_t)b * N_ * 2 + (size_t)(lane & 15) * 2 + half;

    int          count    = (lane < 16) ? 0 : NS_;   // hi lanes: sentinel "full"
    int          firstIdx = 0;
    const size_t qOut     = ((size_t)b * NP_ + qIdx) * NS_;

    float2 cur = pl[0];                   // tile 0 in flight / in regs
    for (int base = 0; base < N_; base += 16) {
        const bool has_next = (base + 16) < N_;
        float2 nxt = cur;
        if (has_next) nxt = pl[(size_t)(base + 16) * 2];   // issue next tile's load

        v2f bop;
        bop.x = cur.x;
        bop.y = cur.y;
        v8f d = __builtin_amdgcn_wmma_f32_16x16x4_f32(
            false, a, false, bop, (short)0, cacc, false, false);

        // Row masks: D VGPR r covers row r (lanes 0-15) and row r+8 (lanes 16-31).
        unsigned sel = 0u;
#pragma unroll
        for (int r = 0; r < 8; ++r) {
            unsigned mk = __builtin_amdgcn_ballot_w32(d[r] < R2_);
            if ((lane & 7) == r) sel = mk;   // lane m<8 -> mask_m ; 8<=m<16 -> mask_{m-8}
        }
        unsigned bits = (lane < 8) ? (sel & 0xFFFFu) : (sel >> 16);
        if (lane >= 16) bits = 0u;

        // Ordered append: first-32-by-index == reference top_k(-sortkey) result.
        while (bits && count < NS_) {
            int bp = __builtin_ctz(bits);
            bits &= bits - 1u;
            int gi = base + bp;
            if (count == 0) firstIdx = gi;
            idxbuf[qOut + count] = gi;
            ++count;
        }

        // Uniform early exit once every owned row has NS_ samples.
        if (__builtin_amdgcn_ballot_w32(count < NS_) == 0u) break;
        cur = nxt;
    }

    // Pad: slots >= count replicate first hit; zero rows get index 0 (matches ref).
    if (lane < 16) {
        for (int s = count; s < NS_; ++s) idxbuf[qOut + s] = firstIdx;
    }
}

// ---------------------------------------------------------------------------
// Kernel C: gather + center xyz + concat features + (0,3,1,2) transpose.
// One thread per (b, np, ns): writes along ns are contiguous 128B per channel.
// Output is write-once -> nontemporal stores keep WGP$/L2 for the reused
// feature rows (each point's 256B row is gathered ~8x on average).
// ---------------------------------------------------------------------------
__global__ __launch_bounds__(256) void gather_concat_kernel(
    const float*  __restrict__ xyz,
    const float*  __restrict__ new_xyz,
    const float4* __restrict__ feat4,
    const int*    __restrict__ idxbuf,
    float*        __restrict__ out) {
    int tid = blockIdx.x * blockDim.x + threadIdx.x;   // over B*NP*NS
    int ns  = tid & (NS_ - 1);
    int np  = (tid >> 5) & (NP_ - 1);
    int b   = tid >> 16;                               // / (NP_*NS_) = 65536
    int pi  = __builtin_nontemporal_load(idxbuf + tid);

    const float* p  = xyz + ((size_t)b * N_ + pi) * 3;
    const float* qc = new_xyz + ((size_t)b * NP_ + np) * 3;
    const size_t cs = (size_t)NP_ * NS_;
    float* ob = out + (size_t)b * (3 + C_) * cs + (size_t)np * NS_ + ns;

    __builtin_nontemporal_store(p[0] - qc[0], ob);
    __builtin_nontemporal_store(p[1] - qc[1], ob + cs);
    __builtin_nontemporal_store(p[2] - qc[2], ob + 2 * cs);

    const float4* f = feat4 + ((size_t)b * N_ + pi) * (C_ / 4);
#pragma unroll
    for (int j = 0; j < C_ / 4; ++j) {
        float4 v  = f[j];
        float* oc = ob + (size_t)(3 + 4 * j) * cs;
        __builtin_nontemporal_store(v.x, oc);
        __builtin_nontemporal_store(v.y, oc + cs);
        __builtin_nontemporal_store(v.z, oc + 2 * cs);
        __builtin_nontemporal_store(v.w, oc + 3 * cs);
    }
}

extern "C" void kernel_launch(void* const* d_in, const int* in_sizes, int n_in,
                              void* d_out, int out_size, void* d_ws, size_t ws_size,
                              hipStream_t stream) {
    const float* xyz      = (const float*)d_in[0];   // (8, 8192, 3)
    const float* new_xyz  = (const float*)d_in[1];   // (8, 2048, 3)
    const float* features = (const float*)d_in[2];   // (8, 8192, 64)
    float*       out      = (float*)d_out;           // (8, 67, 2048, 32)

    float4* pp     = (float4*)d_ws;                                        // 1 MB
    int*    idxbuf = (int*)((char*)d_ws + (size_t)B_ * N_ * sizeof(float4)); // 2 MB

    hipLaunchKernelGGL(pack_points_kernel,
                       dim3((B_ * N_ + 255) / 256), dim3(256), 0, stream,
                       xyz, pp);

    hipLaunchKernelGGL(ball_query_wmma_kernel,
                       dim3(NP_ / 64, B_), dim3(128), 0, stream,
                       new_xyz, (const float2*)pp, idxbuf);

    hipLaunchKernelGGL(gather_concat_kernel,
                       dim3((B_ * NP_ * NS_) / 256), dim3(256), 0, stream,
                       xyz, new_xyz, (const float4*)features, idxbuf, out);
}